// RGATLayer_63857573757392
// MI455X (gfx1250) — compile-verified
//
#include <hip/hip_runtime.h>
#include <hip/hip_bf16.h>
#include <float.h>

// ---------------------------------------------------------------------------
// RGAT layer for MI455X (gfx1250, wave32).
//
// Strategy (roofline-driven):
//  * Exploit basis decomposition: never materialize xp [N,32,64] (164MB).
//    Instead compute xp0 [N, B*U=256] (20.5MB, L2-resident) with one WMMA
//    GEMM, and reconstruct xp[src,rel,h,:] per edge with 4 FMAs/elem.
//  * q/k fold into a second WMMA GEMM: qk[N,64] = xp0 @ Bm1, where
//    Bm1[(b,u), rh]      = coeff[rh,b]*attn_q[rh,u]   (cols 0..31)
//    Bm1[(b,u), 32+rh]   = coeff[rh,b]*attn_k[rh,u]   (cols 32..63)
//  * Segment softmax: f32 atomic max / atomic add into per-(node,head) slots.
//  * Scatter: one wave32 per edge; head-mean (x0.25) folded into the
//    per-(dst,u) atomicAdd so output accumulates directly in d_out.
// ---------------------------------------------------------------------------

#define F_IN   256
#define R_REL  8
#define H_HEADS 4
#define U_DIM  64
#define B_BASIS 4
#define LRELU  0.01f

typedef __attribute__((ext_vector_type(2))) float v2f;
typedef __attribute__((ext_vector_type(8))) float v8f;

// ---- init: out = 0, maxv = -FLT_MAX, denom = 0 -----------------------------
__global__ void k_init(float* __restrict__ out, float* __restrict__ maxv,
                       float* __restrict__ denom, int n_out, int n_nh) {
    int t = blockIdx.x * blockDim.x + threadIdx.x;
    if (t < n_out) out[t] = 0.0f;
    if (t < n_nh) { maxv[t] = -FLT_MAX; denom[t] = 0.0f; }
}

// ---- prep Bm0[f, b*64+u] = basis[b, f, u]  (K-major for GEMM B operand) ----
__global__ void k_prep_b0(const float* __restrict__ basis, float* __restrict__ Bm0) {
    int t = blockIdx.x * blockDim.x + threadIdx.x;      // 256*256
    if (t >= F_IN * B_BASIS * U_DIM) return;
    int f = t >> 8, c = t & 255;
    int b = c >> 6, u = c & 63;
    Bm0[t] = basis[((size_t)b * F_IN + f) * U_DIM + u];
}

// ---- prep Bm1[(b*64+u), col]: col<32 -> q weights, col>=32 -> k weights ----
__global__ void k_prep_b1(const float* __restrict__ coeff, const float* __restrict__ aq,
                          const float* __restrict__ ak, float* __restrict__ Bm1) {
    int t = blockIdx.x * blockDim.x + threadIdx.x;      // 256*64
    if (t >= F_IN * 64) return;
    int f = t >> 6, col = t & 63;
    int b = f >> 6, u = f & 63;
    int rh = col & 31;
    float cf = coeff[rh * B_BASIS + b];
    float av = (col < 32) ? aq[rh * U_DIM + u] : ak[rh * U_DIM + u];
    Bm1[t] = cf * av;
}

// ---- f32 WMMA GEMM: C[M,Nc] = A[M,K] @ Bm[K,Nc]; one wave per 16x16 tile ---
__global__ __launch_bounds__(256)
void k_gemm_wmma(const float* __restrict__ A, const float* __restrict__ Bm,
                 float* __restrict__ C, int M, int Nc, int K) {
    int wave = (blockIdx.x * blockDim.x + threadIdx.x) >> 5;
    int nTilesN = Nc >> 4;
    int totalTiles = (M >> 4) * nTilesN;
    if (wave >= totalTiles) return;
    int m0 = (wave / nTilesN) << 4;
    int n0 = (wave % nTilesN) << 4;

    int lane = threadIdx.x & 31;
    int half = lane >> 4;          // 0: K=0,1   1: K=2,3 (per 32-bit layouts)
    int r    = lane & 15;          // A: row M ; B/C: column N

    v8f acc = {};
    const float* Arow = A + (size_t)(m0 + r) * K;
    #pragma unroll 8
    for (int kb = 0; kb < K; kb += 4) {
        int k0 = kb + 2 * half;
        v2f a, b;
        a.x = Arow[k0];
        a.y = Arow[k0 + 1];
        b.x = Bm[(size_t)k0 * Nc + n0 + r];
        b.y = Bm[(size_t)(k0 + 1) * Nc + n0 + r];
        acc = __builtin_amdgcn_wmma_f32_16x16x4_f32(
            /*neg_a=*/false, a, /*neg_b=*/false, b,
            /*c_mod=*/(short)0, acc, /*reuse_a=*/false, /*reuse_b=*/false);
    }
    // C/D layout: VGPR j -> row = 8*half + j, col = r
    #pragma unroll
    for (int j = 0; j < 8; ++j)
        C[(size_t)(m0 + half * 8 + j) * Nc + n0 + r] = acc[j];
}

// ---- pass 1: per-(edge,head) logit + segment max over dst ------------------
__global__ void k_edge_logit(const float* __restrict__ qk, const int* __restrict__ src,
                             const int* __restrict__ dst, const int* __restrict__ rel,
                             float* __restrict__ elog, float* __restrict__ maxv, int EH) {
    int t = blockIdx.x * blockDim.x + threadIdx.x;
    if (t >= EH) return;
    int e = t >> 2, h = t & 3;
    int s = src[e], d = dst[e], rr = rel[e];
    float q = qk[(size_t)s * 64 + rr * H_HEADS + h];
    float k = qk[(size_t)d * 64 + 32 + rr * H_HEADS + h];
    float z = q + k;
    float l = (z > 0.0f) ? z : LRELU * z;
    elog[t] = l;
    atomicMax(&maxv[d * H_HEADS + h], l);   // global_atomic_max_num_f32
}

// ---- pass 2: ex = exp(logit - max); segment sum over dst -------------------
__global__ void k_edge_exp(const int* __restrict__ dst, const float* __restrict__ maxv,
                           float* __restrict__ elog, float* __restrict__ denom, int EH) {
    int t = blockIdx.x * blockDim.x + threadIdx.x;
    if (t >= EH) return;
    int e = t >> 2, h = t & 3;
    int d = dst[e];
    float ex = __expf(elog[t] - maxv[d * H_HEADS + h]);
    elog[t] = ex;
    atomicAdd(&denom[d * H_HEADS + h], ex);
}

// ---- pass 3: wave32 per edge; reconstruct xp from xp0; scatter-accumulate --
__global__ __launch_bounds__(256)
void k_edge_scatter(const float* __restrict__ xp0, const float* __restrict__ elog,
                    const float* __restrict__ denom, const float* __restrict__ coeff,
                    const int* __restrict__ src, const int* __restrict__ dst,
                    const int* __restrict__ rel, float* __restrict__ out, int E) {
    int e = (blockIdx.x * blockDim.x + threadIdx.x) >> 5;
    int lane = threadIdx.x & 31;
    if (e >= E) return;
    int s = src[e], d = dst[e], rr = rel[e];

    float alpha[H_HEADS], cw[H_HEADS][B_BASIS];
    #pragma unroll
    for (int h = 0; h < H_HEADS; ++h) {
        float den = denom[d * H_HEADS + h];
        alpha[h] = 0.25f * elog[(size_t)e * H_HEADS + h] / fmaxf(den, 1e-16f);
        #pragma unroll
        for (int b = 0; b < B_BASIS; ++b)
            cw[h][b] = coeff[(rr * H_HEADS + h) * B_BASIS + b];
    }
    const float* xrow = xp0 + (size_t)s * (B_BASIS * U_DIM);
    #pragma unroll
    for (int uu = 0; uu < 2; ++uu) {
        int u = lane + uu * 32;
        float xv[B_BASIS];
        #pragma unroll
        for (int b = 0; b < B_BASIS; ++b) xv[b] = xrow[b * U_DIM + u];
        float acc = 0.0f;
        #pragma unroll
        for (int h = 0; h < H_HEADS; ++h) {
            float m = 0.0f;
            #pragma unroll
            for (int b = 0; b < B_BASIS; ++b) m = fmaf(cw[h][b], xv[b], m);
            acc = fmaf(alpha[h], m, acc);
        }
        atomicAdd(&out[(size_t)d * U_DIM + u], acc);
    }
}

extern "C" void kernel_launch(void* const* d_in, const int* in_sizes, int n_in,
                              void* d_out, int out_size, void* d_ws, size_t ws_size,
                              hipStream_t stream) {
    const float* x     = (const float*)d_in[0];
    const float* basis = (const float*)d_in[1];
    const float* coeff = (const float*)d_in[2];
    const float* aq    = (const float*)d_in[3];
    const float* ak    = (const float*)d_in[4];
    const int*   src   = (const int*)d_in[5];
    const int*   dst   = (const int*)d_in[6];
    const int*   rel   = (const int*)d_in[7];
    float* out = (float*)d_out;

    const int N = in_sizes[0] / F_IN;      // 20000
    const int E = in_sizes[5];             // 320000

    // workspace carve-up (floats); total ~31.4 MB
    float* ws   = (float*)d_ws;
    float* Bm0  = ws;                                  // 256*256
    float* Bm1  = Bm0 + F_IN * (B_BASIS * U_DIM);      // 256*64
    float* xp0  = Bm1 + F_IN * 64;                     // N*256
    float* qk   = xp0 + (size_t)N * 256;               // N*64
    float* maxv = qk  + (size_t)N * 64;                // N*4
    float* den  = maxv + (size_t)N * H_HEADS;          // N*4
    float* elog = den  + (size_t)N * H_HEADS;          // E*4

    const int TPB = 256;
    // init out / maxv / denom
    {
        int n_out = N * U_DIM, n_nh = N * H_HEADS;
        int n = n_out > n_nh ? n_out : n_nh;
        k_init<<<(n + TPB - 1) / TPB, TPB, 0, stream>>>(out, maxv, den, n_out, n_nh);
    }
    // weight prep
    k_prep_b0<<<(F_IN * 256 + TPB - 1) / TPB, TPB, 0, stream>>>(basis, Bm0);
    k_prep_b1<<<(F_IN * 64 + TPB - 1) / TPB, TPB, 0, stream>>>(coeff, aq, ak, Bm1);
    // GEMM 1: xp0[N,256] = x[N,256] @ Bm0[256,256]   (WMMA f32)
    {
        int tiles = (N / 16) * (256 / 16);
        k_gemm_wmma<<<(tiles + 7) / 8, TPB, 0, stream>>>(x, Bm0, xp0, N, 256, F_IN);
    }
    // GEMM 2: qk[N,64] = xp0[N,256] @ Bm1[256,64]    (WMMA f32)
    {
        int tiles = (N / 16) * (64 / 16);
        k_gemm_wmma<<<(tiles + 7) / 8, TPB, 0, stream>>>(xp0, Bm1, qk, N, 64, 256);
    }
    // edge passes
    int EH = E * H_HEADS;
    k_edge_logit<<<(EH + TPB - 1) / TPB, TPB, 0, stream>>>(qk, src, dst, rel, elog, maxv, EH);
    k_edge_exp<<<(EH + TPB - 1) / TPB, TPB, 0, stream>>>(dst, maxv, elog, den, EH);
    k_edge_scatter<<<(E * 32 + TPB - 1) / TPB, TPB, 0, stream>>>(
        xp0, elog, den, coeff, src, dst, rel, out, E);
}